// Model_42803644072533
// MI455X (gfx1250) — compile-verified
//
#include <hip/hip_runtime.h>
#include <cstdint>
#include <cstddef>

#define NEG_SLOPE 0.2f

typedef float v2f __attribute__((ext_vector_type(2)));
typedef float v8f __attribute__((ext_vector_type(8)));

// ---------------------------------------------------------------------------
// GEMM: C[M,N] = A[M,K] @ B[K,N], fp32, via V_WMMA_F32_16X16X4_F32.
// One workgroup (256 thr = 8 waves) handles one 16-row M tile; A tile staged
// in LDS; wave w computes N-tiles w, w+8, ... K accumulated in steps of 4.
//
// ISA layouts (CDNA5 ISA):
//   A 16x4 f32 : lane l holds row m=l&15; v0 = A[m][2*(l>>4)], v1 = A[m][2*(l>>4)+1]
//   B  4x16 f32: lane l holds col n=l&15; v0 = B[2*(l>>4)][n], v1 = B[2*(l>>4)+1][n]
//   C/D 16x16  : vgpr r, lanes 0-15 -> (row r,  col lane), lanes 16-31 -> (row r+8, col lane-16)
// ---------------------------------------------------------------------------
__global__ __launch_bounds__(256) void gemm16_wmma_f32(
    const float* __restrict__ A, const float* __restrict__ B,
    float* __restrict__ C, int M, int K, int N)
{
    extern __shared__ float As[];          // 16 * K floats
    const int block_m = blockIdx.x * 16;
    const int tid = threadIdx.x;
    for (int i = tid; i < 16 * K; i += 256) {
        int r = i / K, c = i - r * K;
        int row = block_m + r;
        As[i] = (row < M) ? A[(size_t)row * K + c] : 0.0f;
    }
    __syncthreads();

    const int wave  = tid >> 5;
    const int lane  = tid & 31;
    const int m16   = lane & 15;
    const int khalf = (lane >> 4) << 1;    // 0 or 2
    const int ntiles = N >> 4;

    for (int nt = wave; nt < ntiles; nt += 8) {
        const int ncol = (nt << 4) + m16;
        v8f acc = {};
        for (int k = 0; k < K; k += 4) {
            v2f a, b;
            a.x = As[m16 * K + k + khalf];
            a.y = As[m16 * K + k + khalf + 1];
            b.x = B[(size_t)(k + khalf) * N + ncol];
            b.y = B[(size_t)(k + khalf + 1) * N + ncol];
            acc = __builtin_amdgcn_wmma_f32_16x16x4_f32(
                false, a, false, b, (short)0, acc, false, false);
        }
        const int rbase = block_m + ((lane >> 4) << 3);
#pragma unroll
        for (int r = 0; r < 8; ++r) {
            int row = rbase + r;
            if (row < M) C[(size_t)row * N + ncol] = acc[r];
        }
    }
}

// ---------------------------------------------------------------------------
// Per-node attention logits: as[n,h] = sum_c h[n,h,c]*a_src[h,c], same for ad.
// ---------------------------------------------------------------------------
__global__ void alpha_kernel(const float* __restrict__ h,
                             const float* __restrict__ a_src,
                             const float* __restrict__ a_dst,
                             float* __restrict__ as_o, float* __restrict__ ad_o,
                             int N, int H, int C)
{
    int i = blockIdx.x * blockDim.x + threadIdx.x;   // n*H + head
    if (i >= N * H) return;
    int n = i / H, hh = i - n * H;
    const float* hp  = h + (size_t)n * H * C + (size_t)hh * C;
    const float* sp  = a_src + hh * C;
    const float* dp  = a_dst + hh * C;
    float s = 0.f, d = 0.f;
    for (int c = 0; c < C; ++c) { s += hp[c] * sp[c]; d += hp[c] * dp[c]; }
    as_o[i] = s; ad_o[i] = d;
}

// Monotonic float <-> uint key for atomic max over possibly-negative floats.
__device__ __forceinline__ unsigned fkey(float f) {
    unsigned u = __float_as_uint(f);
    return (u & 0x80000000u) ? ~u : (u | 0x80000000u);
}
__device__ __forceinline__ float funkey(unsigned u) {
    unsigned b = (u & 0x80000000u) ? (u ^ 0x80000000u) : ~u;
    return __uint_as_float(b);
}

// e = leakyrelu(as[src]+ad[dst]); segment max over dst via atomicMax on keys.
__global__ void edge_score_kernel(const int* __restrict__ ei, int E, int Etot,
                                  const float* __restrict__ as_i,
                                  const float* __restrict__ ad_i,
                                  float* __restrict__ ebuf,
                                  unsigned* __restrict__ maxk, int H)
{
    long long i = (long long)blockIdx.x * blockDim.x + threadIdx.x;
    if (i >= (long long)Etot * H) return;
    int e = (int)(i / H), hh = (int)(i - (long long)e * H);
    int s = (e < E) ? ei[e]     : (e - E);
    int d = (e < E) ? ei[E + e] : (e - E);
    float v = as_i[s * H + hh] + ad_i[d * H + hh];
    v = (v > 0.f) ? v : NEG_SLOPE * v;
    ebuf[i] = v;
    atomicMax(&maxk[d * H + hh], fkey(v));
}

// ee = exp(e - max[dst]); segment sum into denom.
__global__ void edge_exp_kernel(const int* __restrict__ ei, int E, int Etot,
                                float* __restrict__ ebuf,
                                const unsigned* __restrict__ maxk,
                                float* __restrict__ denom, int H)
{
    long long i = (long long)blockIdx.x * blockDim.x + threadIdx.x;
    if (i >= (long long)Etot * H) return;
    int e = (int)(i / H), hh = (int)(i - (long long)e * H);
    int d = (e < E) ? ei[E + e] : (e - E);
    float m  = funkey(maxk[d * H + hh]);
    float ee = expf(ebuf[i] - m);
    ebuf[i] = ee;
    atomicAdd(&denom[d * H + hh], ee);
}

// acc[dst,h,c] += h[src,h,c] * ee/(denom[dst,h]+eps); 4 channels per thread.
__global__ void scatter_kernel(const int* __restrict__ ei, int E, int Etot,
                               const float* __restrict__ h,
                               const float* __restrict__ ebuf,
                               const float* __restrict__ denom,
                               float* __restrict__ acc, int H, int C)
{
    const int HC = H * C;
    const int G  = HC >> 2;
    long long i = (long long)blockIdx.x * blockDim.x + threadIdx.x;
    if (i >= (long long)Etot * G) return;
    int e = (int)(i / G);
    int g = (int)(i - (long long)e * G);
    int c  = g << 2;
    int hh = c / C;
    int s = (e < E) ? ei[e]     : (e - E);
    int d = (e < E) ? ei[E + e] : (e - E);
    float alpha = ebuf[(size_t)e * H + hh] / (denom[d * H + hh] + 1e-16f);
    const float4 hv = *(const float4*)(h + (size_t)s * HC + c);
    float* ap = acc + (size_t)d * HC + c;
    atomicAdd(ap + 0, hv.x * alpha);
    atomicAdd(ap + 1, hv.y * alpha);
    atomicAdd(ap + 2, hv.z * alpha);
    atomicAdd(ap + 3, hv.w * alpha);
}

// out = (relu?)(acc + bias), in place.
__global__ void bias_act_kernel(float* __restrict__ acc, const float* __restrict__ b,
                                int total, int HC, int do_relu)
{
    int i = blockIdx.x * blockDim.x + threadIdx.x;
    if (i >= total) return;
    float v = acc[i] + b[i % HC];
    if (do_relu) v = fmaxf(v, 0.f);
    acc[i] = v;
}

// out[n,c] = mean_h acc[n,h,c] + b[c]
__global__ void head_mean_kernel(const float* __restrict__ acc,
                                 const float* __restrict__ b,
                                 float* __restrict__ out, int N, int H, int C)
{
    int i = blockIdx.x * blockDim.x + threadIdx.x;   // n*C + c
    if (i >= N * C) return;
    int n = i / C, c = i - n * C;
    float s = 0.f;
    for (int hh = 0; hh < H; ++hh) s += acc[(size_t)n * H * C + (size_t)hh * C + c];
    out[i] = s / (float)H + b[c];
}

// ---------------------------------------------------------------------------
extern "C" void kernel_launch(void* const* d_in, const int* in_sizes, int n_in,
                              void* d_out, int out_size, void* d_ws, size_t ws_size,
                              hipStream_t stream)
{
    if (n_in < 14) return;
    const float* x   = (const float*)d_in[0];
    const int*   ei  = (const int*)  d_in[1];   // [2,E] row-major: src then dst
    const float* W1  = (const float*)d_in[2];
    const float* as1 = (const float*)d_in[3];
    const float* ad1 = (const float*)d_in[4];
    const float* b1  = (const float*)d_in[5];
    const float* W2  = (const float*)d_in[6];
    const float* as2 = (const float*)d_in[7];
    const float* ad2 = (const float*)d_in[8];
    const float* b2  = (const float*)d_in[9];
    const float* W3  = (const float*)d_in[10];
    const float* as3 = (const float*)d_in[11];
    const float* ad3 = (const float*)d_in[12];
    const float* b3  = (const float*)d_in[13];

    const int IN   = 128;                 // input feature dim
    const int N    = in_sizes[0] / IN;    // nodes
    const int E    = in_sizes[1] / 2;     // edges (before self-loops)
    const int Etot = E + N;               // + self loops
    const int H12 = 8,  C12 = 16;         // layers 1,2: 8 heads x 16
    const int H3  = 6,  C3  = 40;         // layer 3:    6 heads x 40
    const int HCmax = H3 * C3;            // 240 >= 128

    // Workspace layout (floats)
    float*    HB   = (float*)d_ws;                       // [N, HCmax] h = X@W
    float*    ACC  = HB  + (size_t)N * HCmax;            // [N, HCmax] scatter acc / next X
    float*    EB   = ACC + (size_t)N * HCmax;            // [Etot, 8] edge scores/ee
    float*    AS   = EB  + (size_t)Etot * 8;             // [N, 8]
    float*    AD   = AS  + (size_t)N * 8;                // [N, 8]
    unsigned* MAXK = (unsigned*)(AD + (size_t)N * 8);    // [N, 8]
    float*    DEN  = (float*)(MAXK + (size_t)N * 8);     // [N, 8]

    auto run_layer = [&](const float* Xin, int K, const float* W,
                         const float* a_s, const float* a_d, int H, int C) {
        const int HC = H * C;
        // 1) h = Xin @ W  (WMMA f32)
        gemm16_wmma_f32<<<(N + 15) / 16, 256, 16 * K * sizeof(float), stream>>>(
            Xin, W, HB, N, K, HC);
        // 2) per-node logits
        int nh = N * H;
        alpha_kernel<<<(nh + 255) / 256, 256, 0, stream>>>(HB, a_s, a_d, AS, AD, N, H, C);
        // 3) clear reductions
        hipMemsetAsync(MAXK, 0, (size_t)N * H * sizeof(unsigned), stream);
        hipMemsetAsync(DEN,  0, (size_t)N * H * sizeof(float),    stream);
        hipMemsetAsync(ACC,  0, (size_t)N * HC * sizeof(float),   stream);
        // 4) edge scores + segment max
        long long eh = (long long)Etot * H;
        int gb = (int)((eh + 255) / 256);
        edge_score_kernel<<<gb, 256, 0, stream>>>(ei, E, Etot, AS, AD, EB, MAXK, H);
        // 5) exp + segment sum
        edge_exp_kernel<<<gb, 256, 0, stream>>>(ei, E, Etot, EB, MAXK, DEN, H);
        // 6) weighted scatter of messages
        long long eg = (long long)Etot * (HC >> 2);
        scatter_kernel<<<(int)((eg + 255) / 256), 256, 0, stream>>>(
            ei, E, Etot, HB, EB, DEN, ACC, H, C);
    };

    // Layer 1: x -> ACC = relu(gat + b1)
    run_layer(x, IN, W1, as1, ad1, H12, C12);
    bias_act_kernel<<<(N * 128 + 255) / 256, 256, 0, stream>>>(ACC, b1, N * 128, 128, 1);
    // Layer 2: ACC -> ACC = gat + b2
    run_layer(ACC, 128, W2, as2, ad2, H12, C12);
    bias_act_kernel<<<(N * 128 + 255) / 256, 256, 0, stream>>>(ACC, b2, N * 128, 128, 0);
    // Layer 3: ACC -> out = mean_heads(gat) + b3
    run_layer(ACC, 128, W3, as3, ad3, H3, C3);
    head_mean_kernel<<<(N * C3 + 255) / 256, 256, 0, stream>>>(
        ACC, b3, (float*)d_out, N, H3, C3);
    (void)out_size; (void)ws_size;
}